// Encoder_eCHOLGA_42356967473564
// MI455X (gfx1250) — compile-verified
//
#include <hip/hip_runtime.h>
#include <hip/hip_bf16.h>
#include <cfloat>

// ---------------------------------------------------------------------------
// HGT (heterogeneous graph transformer), 2 layers, for MI455X (gfx1250).
// GEMMs use V_WMMA_F32_16X16X4_F32 (full fp32 fidelity on the matrix pipe,
// wave32 fragment layouts per CDNA5 ISA 7.12.2). Tile staging uses the CDNA5
// async global->LDS path (GLOBAL_LOAD_ASYNC_TO_LDS_B128, ASYNCcnt) instead of
// a VGPR round-trip. Edge attention stages are wave-per-edge coalesced with
// L2 float atomics.
// ---------------------------------------------------------------------------

typedef __attribute__((ext_vector_type(2))) float v2f;
typedef __attribute__((ext_vector_type(8))) float v8f;

constexpr int Nn  = 20000;   // nodes per type
constexpr int NEe = 80000;   // edges per edge type
constexpr int Tt  = 6;       // node types
constexpr int Ee  = 10;      // edge types
constexpr int INf = 256;
constexpr int H1  = 128;
constexpr int H2  = 64;
constexpr int Cc  = 6;

__constant__ int c_SRC_T[10] = {1, 2, 3, 4, 5, 0, 0, 0, 0, 0};
__constant__ int c_DST_T[10] = {0, 0, 0, 0, 0, 1, 2, 3, 4, 5};

// Async 16-byte global->LDS copy, GVS mode (SGPR 64b base + VGPR 32b offset).
// LDS destination address passed in the VDST VGPR. Tracked by ASYNCcnt.
__device__ __forceinline__ void async_copy_b128(const void* lds_ptr,
                                                const float* sbase,
                                                unsigned goff_bytes)
{
    unsigned loff = (unsigned)(uintptr_t)lds_ptr;   // low 32 bits = LDS offset
    asm volatile("global_load_async_to_lds_b128 %0, %1, %2"
                 :
                 : "v"(loff), "v"(goff_bytes), "s"(sbase)
                 : "memory");
}

__device__ __forceinline__ void async_wait_all()
{
    asm volatile("s_wait_asynccnt 0" ::: "memory");
}

// ---------------------------------------------------------------------------
// Generic batched GEMM:  C[bz] = A[map(bz)] @ B[bz] (+ bias[bz])
// Block: 256 threads = 8 waves -> 32x64 output tile (2x4 waves of 16x16).
// K staged through LDS in chunks of 64 via async B128 copies, then a
// WMMA f32 16x16x4 inner loop off LDS.
// Requires: M % 32 == 0, Nc % 64 == 0, K % 64 == 0 (all true here).
// ---------------------------------------------------------------------------
template <int KT>
__global__ __launch_bounds__(256)
void gemm_wmma_f32(const float* __restrict__ A, int lda, long long aStr, int aMap,
                   const float* __restrict__ B, int ldb, long long bStr,
                   const float* __restrict__ bias, int biasStr,
                   float* __restrict__ Cmat, int ldc, long long cStr)
{
    const int bz = blockIdx.z;
    const int ab = aMap ? c_SRC_T[bz] : bz;
    const float* Ab = A + (long long)ab * aStr;
    const float* Bb = B + (long long)bz * bStr;
    float*       Cb = Cmat + (long long)bz * cStr;

    const int mBlock = blockIdx.x * 32;
    const int nBlock = blockIdx.y * 64;

    const int tid  = threadIdx.x;
    const int wave = tid >> 5;
    const int lane = tid & 31;
    const int wm   = wave >> 2;    // 0..1
    const int wn   = wave & 3;     // 0..3

    // +4 float pad: rows stay 16B-aligned for B128 async copies (stride 272B),
    // and the 16-lane column reads of the A fragment hit distinct banks
    // (bank stride = 68 mod 64 = 4).
    __shared__ float As[32][68];
    __shared__ float Bs[64][68];

    v8f acc = {};
    const int r  = lane & 15;
    const int hk = (lane >> 4) * 2;      // K sub-pair select (ISA 7.12.2)
    const int rl = wm * 16 + r;          // local A row
    const int cl = wn * 16 + r;          // local B/C col

    for (int kb = 0; kb < KT; kb += 64) {
        // Stage A chunk: 32 x 64 floats = 512 x 16B, 2 async B128 per thread
#pragma unroll
        for (int i = 0; i < 2; ++i) {
            int idx = tid + i * 256;         // 0..511
            int rr  = idx >> 4;              // 0..31
            int c4  = (idx & 15) * 4;        // 0..60, 16B groups
            unsigned goff = (unsigned)((((long long)(mBlock + rr)) * lda + (kb + c4)) * 4);
            async_copy_b128(&As[rr][c4], Ab, goff);
        }
        // Stage B chunk: 64 x 64 floats = 1024 x 16B, 4 async B128 per thread
#pragma unroll
        for (int i = 0; i < 4; ++i) {
            int idx = tid + i * 256;         // 0..1023
            int rr  = idx >> 4;              // 0..63
            int c4  = (idx & 15) * 4;
            unsigned goff = (unsigned)((((long long)(kb + rr)) * ldb + (nBlock + c4)) * 4);
            async_copy_b128(&Bs[rr][c4], Bb, goff);
        }
        async_wait_all();      // our wave's async copies have landed in LDS
        __syncthreads();       // all waves' copies visible

#pragma unroll
        for (int kc = 0; kc < 64; kc += 4) {
            v2f a, b;
            a.x = As[rl][kc + hk];
            a.y = As[rl][kc + hk + 1];
            b.x = Bs[kc + hk][cl];
            b.y = Bs[kc + hk + 1][cl];
            // D = A(16x4) * B(4x16) + C, fp32 matrix pipe
            acc = __builtin_amdgcn_wmma_f32_16x16x4_f32(
                false, a, false, b, (short)0, acc, false, false);
        }
        __syncthreads();
    }

    const int col   = nBlock + cl;
    const float bv  = bias ? bias[(long long)bz * biasStr + col] : 0.0f;
    const int rbase = mBlock + wm * 16 + (lane >> 4) * 8;
#pragma unroll
    for (int j = 0; j < 8; ++j)
        Cb[(long long)(rbase + j) * ldc + col] = acc[j] + bv;
}

// ---------------------------------------------------------------------------
// Utility kernels
// ---------------------------------------------------------------------------
__global__ void fill_kernel(float* __restrict__ p, float v, long long n)
{
    long long i = (long long)blockIdx.x * blockDim.x + threadIdx.x;
    if (i < n) p[i] = v;
}

__device__ __forceinline__ void atomicMaxF(float* addr, float val)
{
    unsigned int* ua = (unsigned int*)addr;
    unsigned int old = *ua;
    while (true) {
        float cur = __uint_as_float(old);
        if (cur >= val) break;
        unsigned int assumed = old;
        old = atomicCAS(ua, assumed, __float_as_uint(val));
        if (old == assumed) break;
    }
}

// alpha[e,j] = (q[DST_T[e]][dst] . ktr[e][src]) * p[e] / sqrt(H); segmax via atomic
__global__ __launch_bounds__(256)
void edge_scores(const float* __restrict__ kqv, int ldkqv, int H,
                 const float* __restrict__ ktr,
                 const int* __restrict__ ei,
                 const float* __restrict__ p_rel,
                 float* __restrict__ alpha,
                 float* __restrict__ segmax)
{
    const int e    = blockIdx.y;
    const int j    = (int)((blockIdx.x * blockDim.x + threadIdx.x) >> 5);
    const int lane = threadIdx.x & 31;
    if (j >= NEe) return;

    const int src = ei[(e * 2) * NEe + j];
    const int dst = ei[(e * 2 + 1) * NEe + j];
    const int tq  = c_DST_T[e];

    const float* q = kqv + ((long long)tq * Nn + dst) * ldkqv + H; // q slice
    const float* k = ktr + ((long long)e * Nn + src) * H;

    float acc = 0.0f;
    for (int d = lane; d < H; d += 32) acc += q[d] * k[d];
#pragma unroll
    for (int m = 16; m; m >>= 1) acc += __shfl_xor(acc, m, 32);

    if (lane == 0) {
        float aj = acc * p_rel[e] * rsqrtf((float)H);
        alpha[(long long)e * NEe + j] = aj;
        atomicMaxF(&segmax[(long long)tq * Nn + dst], aj);
    }
}

// w = exp(alpha - segmax); segsum += w (thread per edge)
__global__ __launch_bounds__(256)
void edge_expsum(const int* __restrict__ ei,
                 float* __restrict__ alpha,
                 const float* __restrict__ segmax,
                 float* __restrict__ segsum)
{
    const int idx = blockIdx.x * 256 + threadIdx.x;     // E*NE = 800000 exact
    const int e = idx / NEe, j = idx % NEe;
    const int dst = ei[(e * 2 + 1) * NEe + j];
    const int seg = c_DST_T[e] * Nn + dst;
    float w = __expf(alpha[idx] - segmax[seg]);
    alpha[idx] = w;
    atomicAdd(&segsum[seg], w);
}

// agg[seg] += (w / segsum) * vtr[e][src]  (wave per edge, lane-strided dims)
__global__ __launch_bounds__(256)
void edge_scatter(const int* __restrict__ ei,
                  const float* __restrict__ alpha,
                  const float* __restrict__ segsum,
                  const float* __restrict__ vtr, int H,
                  float* __restrict__ agg)
{
    const int e    = blockIdx.y;
    const int j    = (int)((blockIdx.x * blockDim.x + threadIdx.x) >> 5);
    const int lane = threadIdx.x & 31;
    if (j >= NEe) return;

    const int src = ei[(e * 2) * NEe + j];
    const int dst = ei[(e * 2 + 1) * NEe + j];
    const int seg = c_DST_T[e] * Nn + dst;

    const float coeff = alpha[(long long)e * NEe + j] / (segsum[seg] + 1e-16f);
    const float* v = vtr + ((long long)e * Nn + src) * H;
    float*       o = agg + (long long)seg * H;
    for (int d = lane; d < H; d += 32) atomicAdd(&o[d], coeff * v[d]);
}

__global__ void gelu_kernel(float* __restrict__ x, long long n)
{
    long long i = (long long)blockIdx.x * blockDim.x + threadIdx.x;
    if (i < n) {
        float v = x[i];
        x[i] = 0.5f * v * (1.0f + erff(v * 0.70710678118654752f));
    }
}

// all_rep = tanh(h2); rel = all_rep[0:N*H2]
__global__ void tanh_out_kernel(const float* __restrict__ h2,
                                float* __restrict__ rel,
                                float* __restrict__ all_rep,
                                long long n)
{
    long long i = (long long)blockIdx.x * blockDim.x + threadIdx.x;
    if (i < n) {
        float v = tanhf(h2[i]);
        all_rep[i] = v;
        if (i < (long long)Nn * H2) rel[i] = v;
    }
}

// logits = all_rep(120000x64) @ W3(64x6) + b3 — wave per row
__global__ __launch_bounds__(256)
void logits_kernel(const float* __restrict__ all_rep,
                   const float* __restrict__ W3,
                   const float* __restrict__ b3,
                   float* __restrict__ logits)
{
    const int row  = (int)((blockIdx.x * blockDim.x + threadIdx.x) >> 5);
    const int lane = threadIdx.x & 31;
    if (row >= Tt * Nn) return;
    const float* h = all_rep + (long long)row * H2;
    const float h0 = h[lane], h1 = h[lane + 32];
#pragma unroll
    for (int c = 0; c < Cc; ++c) {
        float p = h0 * W3[lane * Cc + c] + h1 * W3[(lane + 32) * Cc + c];
#pragma unroll
        for (int m = 16; m; m >>= 1) p += __shfl_down(p, m, 32);
        if (lane == 0) logits[(long long)row * Cc + c] = p + b3[c];
    }
}

// ---------------------------------------------------------------------------
// Host-side orchestration
// ---------------------------------------------------------------------------
static void launch_gemm(hipStream_t s, int K, int M, int Ncols, int batches,
                        const float* A, int lda, long long aStr, int aMap,
                        const float* B, int ldb, long long bStr,
                        const float* bias, int biasStr,
                        float* C, int ldc, long long cStr)
{
    dim3 g(M / 32, Ncols / 64, batches), b(256);
    switch (K) {
    case 256: gemm_wmma_f32<256><<<g, b, 0, s>>>(A, lda, aStr, aMap, B, ldb, bStr, bias, biasStr, C, ldc, cStr); break;
    case 128: gemm_wmma_f32<128><<<g, b, 0, s>>>(A, lda, aStr, aMap, B, ldb, bStr, bias, biasStr, C, ldc, cStr); break;
    case  64: gemm_wmma_f32< 64><<<g, b, 0, s>>>(A, lda, aStr, aMap, B, ldb, bStr, bias, biasStr, C, ldc, cStr); break;
    default: break;
    }
}

static void hgt_layer(hipStream_t s,
                      const float* x, int inF, int H,
                      const float* Wkqv, const float* bkqv,
                      const float* a_rel, const float* m_rel, const float* p_rel,
                      const float* Wout, const float* bout,
                      const int* ei,
                      float* kqv, float* ktr, float* vtr,
                      float* alpha, float* segmax, float* segsum,
                      float* agg, float* out)
{
    const int threeH = 3 * H;

    // kqv = x @ Wkqv + bkqv   (T batches, K=inF, Nc=3H)
    launch_gemm(s, inF, Nn, threeH, Tt,
                x, inF, (long long)Nn * inF, 0,
                Wkqv, threeH, (long long)inF * threeH,
                bkqv, threeH,
                kqv, threeH, (long long)Nn * threeH);

    // ktr[e] = k[SRC_T[e]] @ a_rel[e]   (E batches, K=H, Nc=H)
    launch_gemm(s, H, Nn, H, Ee,
                kqv /* k slice, offset 0 */, threeH, (long long)Nn * threeH, 1,
                a_rel, H, (long long)H * H,
                nullptr, 0,
                ktr, H, (long long)Nn * H);

    // vtr[e] = v[SRC_T[e]] @ m_rel[e]
    launch_gemm(s, H, Nn, H, Ee,
                kqv + 2 * H, threeH, (long long)Nn * threeH, 1,
                m_rel, H, (long long)H * H,
                nullptr, 0,
                vtr, H, (long long)Nn * H);

    // segment softmax state + aggregation buffer
    {
        long long nseg = (long long)Tt * Nn;
        long long nagg = nseg * H;
        fill_kernel<<<(unsigned)((nseg + 255) / 256), 256, 0, s>>>(segmax, -FLT_MAX, nseg);
        fill_kernel<<<(unsigned)((nseg + 255) / 256), 256, 0, s>>>(segsum, 0.0f, nseg);
        fill_kernel<<<(unsigned)((nagg + 255) / 256), 256, 0, s>>>(agg, 0.0f, nagg);
    }

    dim3 ge(NEe / 8, Ee);
    edge_scores<<<ge, 256, 0, s>>>(kqv, threeH, H, ktr, ei, p_rel, alpha, segmax);
    edge_expsum<<<(Ee * NEe) / 256, 256, 0, s>>>(ei, alpha, segmax, segsum);
    edge_scatter<<<ge, 256, 0, s>>>(ei, alpha, segsum, vtr, H, agg);

    long long nagg = (long long)Tt * Nn * H;
    gelu_kernel<<<(unsigned)((nagg + 255) / 256), 256, 0, s>>>(agg, nagg);

    // out = gelu(agg) @ Wout + bout   (T batches, K=H, Nc=H)
    launch_gemm(s, H, Nn, H, Tt,
                agg, H, (long long)Nn * H, 0,
                Wout, H, (long long)H * H,
                bout, H,
                out, H, (long long)Nn * H);
}

extern "C" void kernel_launch(void* const* d_in, const int* in_sizes, int n_in,
                              void* d_out, int out_size, void* d_ws, size_t ws_size,
                              hipStream_t stream)
{
    (void)in_sizes; (void)n_in; (void)out_size; (void)ws_size;

    const float* x       = (const float*)d_in[0];
    const int*   ei      = (const int*)  d_in[1];
    const float* Wkqv1   = (const float*)d_in[2];
    const float* bkqv1   = (const float*)d_in[3];
    const float* a1      = (const float*)d_in[4];
    const float* m1      = (const float*)d_in[5];
    const float* p1      = (const float*)d_in[6];
    const float* Wout1   = (const float*)d_in[7];
    const float* bout1   = (const float*)d_in[8];
    const float* Wkqv2   = (const float*)d_in[9];
    const float* bkqv2   = (const float*)d_in[10];
    const float* a2      = (const float*)d_in[11];
    const float* m2      = (const float*)d_in[12];
    const float* p2      = (const float*)d_in[13];
    const float* Wout2   = (const float*)d_in[14];
    const float* bout2   = (const float*)d_in[15];
    const float* W3      = (const float*)d_in[16];
    const float* b3      = (const float*)d_in[17];

    // Workspace layout (floats), buffers reused across the two layers.
    float* ws = (float*)d_ws;
    float* kqv    = ws;                                  // max(T*N*384, T*N*192)
    float* ktr    = kqv    + (long long)Tt * Nn * 384;   // E*N*128
    float* vtr    = ktr    + (long long)Ee * Nn * 128;   // E*N*128
    float* agg    = vtr    + (long long)Ee * Nn * 128;   // T*N*128
    float* hbuf   = agg    + (long long)Tt * Nn * 128;   // T*N*128 (h1 -> h2)
    float* alpha  = hbuf   + (long long)Tt * Nn * 128;   // E*NE
    float* segmax = alpha  + (long long)Ee * NEe;        // T*N
    float* segsum = segmax + (long long)Tt * Nn;         // T*N

    // Layer 1: IN=256 -> H1=128
    hgt_layer(stream, x, INf, H1, Wkqv1, bkqv1, a1, m1, p1, Wout1, bout1,
              ei, kqv, ktr, vtr, alpha, segmax, segsum, agg, hbuf);

    // Layer 2: H1=128 -> H2=64 (reuses kqv/ktr/vtr/agg buffers; out overwrites hbuf
    // after its only use as layer-2 input in the kqv GEMM)
    hgt_layer(stream, hbuf, H1, H2, Wkqv2, bkqv2, a2, m2, p2, Wout2, bout2,
              ei, kqv, ktr, vtr, alpha, segmax, segsum, agg, hbuf);

    // Outputs: rel (N*H2) | all_rep (T*N*H2) | logits (T*N*C)
    float* rel     = (float*)d_out;
    float* all_rep = rel + (long long)Nn * H2;
    float* logits  = all_rep + (long long)Tt * Nn * H2;

    long long ntot = (long long)Tt * Nn * H2;
    tanh_out_kernel<<<(unsigned)((ntot + 255) / 256), 256, 0, stream>>>(hbuf, rel, all_rep, ntot);

    int rows = Tt * Nn;                // 120000 rows, wave per row
    logits_kernel<<<(rows * 32 + 255) / 256, 256, 0, stream>>>(all_rep, W3, b3, logits);
}